// RSA_34823594836498
// MI455X (gfx1250) — compile-verified
//
#include <hip/hip_runtime.h>
#include <hip/hip_bf16.h>

// ---------------------------------------------------------------------------
// Shifted-window channel ("transposed") attention, b=4 c=128 H=W=256 ws=8.
// Memory-bound (~1.3 GB HBM traffic vs 53.5 GFLOP) -> keep fp32, use
// V_WMMA_F32_16X16X4_F32 for all GEMMs. This revision:
//   - K1/K4: B-fragments register-resident, reused across all M-tiles
//   - K2: LDS-staged depthwise conv, fully coalesced windowed writes
//   - K3: async global->LDS fill (ASYNCcnt) + prefetch of v
// ---------------------------------------------------------------------------

typedef float v2f __attribute__((ext_vector_type(2)));
typedef float v8f __attribute__((ext_vector_type(8)));

#define C_      128
#define C3_     384
#define H_      256
#define W_      256
#define B_      4
#define HW_     65536          // 256*256
#define NWIN_   1024           // 32*32 windows per image
#define P_      64             // pixels per window
#define PIX_T_  16384          // (B*H*W)/16 pixel tiles

static __device__ __forceinline__ v8f wmma4(v2f a, v2f b, v8f c) {
  // D = A(16x4, f32) * B(4x16, f32) + C(16x16, f32)
  return __builtin_amdgcn_wmma_f32_16x16x4_f32(false, a, false, b, (short)0, c,
                                               false, false);
}

// ---------------------------------------------------------------------------
// K1: qkv = W_qkv[384x128] @ x(rolled by -4,-4).
// One wave owns a 16-pixel tile: load the 64 B-fragment floats once into
// registers, then sweep all 24 out-channel tiles (A from L0-hot w_qkv).
// qkv_raw layout: [b][384][h][w] (K2 reads 3x3 taps coalesced).
// ---------------------------------------------------------------------------
__global__ void __launch_bounds__(256) k1_qkv_gemm(const float* __restrict__ x,
                                                   const float* __restrict__ wq,
                                                   float* __restrict__ qkv_raw) {
  const int tid  = threadIdx.x;
  const int lane = tid & 31;
  const int wv   = tid >> 5;
  const int pt   = blockIdx.x * 8 + wv;          // 16384 pixel tiles
  const int pixBase = pt * 16;                   // 16 consecutive px, same b,h
  const int b   = pixBase >> 16;
  const int rem = pixBase & (HW_ - 1);
  const int h   = rem >> 8;
  const int wb  = rem & (W_ - 1);

  const int mrel = lane & 15;
  const int kb   = (lane >> 4) << 1;             // 0 or 2
  const int hh   = (h + 4) & (H_ - 1);           // roll(-4): x_[h] = x[(h+4)%H]
  const int ww   = (wb + mrel + 4) & (W_ - 1);
  const float* xb = x + (size_t)b * C_ * HW_ + hh * W_ + ww;

  v2f barr[32];                                  // whole K column, reg-resident
#pragma unroll
  for (int i = 0; i < 32; ++i) {
    barr[i].x = xb[(size_t)(4 * i + kb) * HW_];
    barr[i].y = xb[(size_t)(4 * i + kb + 1) * HW_];
  }

  const int rowHi = (lane >> 4) * 8;
  for (int mt = 0; mt < 24; ++mt) {
    const int m0 = mt * 16;
    v8f acc = {};
#pragma unroll
    for (int i = 0; i < 32; ++i) {
      v2f a;
      a.x = wq[(m0 + mrel) * C_ + 4 * i + kb];
      a.y = wq[(m0 + mrel) * C_ + 4 * i + kb + 1];
      acc = wmma4(a, barr[i], acc);
    }
#pragma unroll
    for (int r = 0; r < 8; ++r)
      qkv_raw[((size_t)(b * C3_ + m0 + rowHi + r)) * HW_ + h * W_ + wb + mrel] =
          acc[r];
  }
}

// ---------------------------------------------------------------------------
// K2: depthwise 3x3 (zero pad 1), LDS-staged.  One block per (b, which, win):
// stage 10x10 halo for 128 channels (51.2KB) + taps, then write the window's
// [p][c] tile fully coalesced into qkvw ([b][win][which][p][c]).
// ---------------------------------------------------------------------------
__global__ void __launch_bounds__(256) k2_dwconv(const float* __restrict__ qkv_raw,
                                                 const float* __restrict__ wdw,
                                                 float* __restrict__ qkvw) {
  __shared__ float halo[128 * 100];              // [c][iy*10+ix], 51.2 KB
  __shared__ float taps[128 * 9];                // 4.6 KB

  const int tid = threadIdx.x;
  const int bi  = blockIdx.x;                    // B*3*NWIN = 12288
  const int b     = bi / (3 * NWIN_);
  const int rem   = bi % (3 * NWIN_);
  const int which = rem / NWIN_;
  const int win   = rem % NWIN_;
  const int wy = win >> 5, wx = win & 31;

  for (int e = tid; e < 128 * 9; e += 256)
    taps[e] = wdw[(which * C_) * 9 + e];

  const float* inb =
      qkv_raw + ((size_t)(b * C3_ + which * C_)) * HW_;
  for (int e = tid; e < 128 * 100; e += 256) {
    const int c = e / 100, i = e % 100;
    const int hy  = wy * 8 + (i / 10) - 1;
    const int wx2 = wx * 8 + (i % 10) - 1;
    float v = 0.f;
    if (hy >= 0 && hy < H_ && wx2 >= 0 && wx2 < W_)
      v = inb[(size_t)c * HW_ + hy * W_ + wx2];
    halo[e] = v;
  }
  __syncthreads();

  float* ob = qkvw + (size_t)((b * NWIN_ + win) * 3 + which) * (P_ * C_);
  for (int e = tid; e < P_ * C_; e += 256) {     // e = p*128 + c  (coalesced)
    const int p = e >> 7, c = e & (C_ - 1);
    const int dy = p >> 3, dx = p & 7;
    const float* hb = &halo[c * 100 + dy * 10 + dx];
    const float* tb = &taps[c * 9];
    float s = 0.f;
#pragma unroll
    for (int ky = 0; ky < 3; ++ky)
#pragma unroll
      for (int kx = 0; kx < 3; ++kx)
        s += hb[ky * 10 + kx] * tb[ky * 3 + kx];
    ob[e] = s;
  }
}

// ---------------------------------------------------------------------------
// K3: per-window attention.  Block = 1 window (256 thr = 8 waves).
//   q,k streamed into LDS with global_load_async_to_lds_b128 (ASYNCcnt),
//   v tile prefetched meanwhile.  attn overlays the q/k LDS after GEMM1.
//   GEMM1: attn = relu(T * q^T k)   (M=c_q, N=c_k, K=pixels)
//   GEMM2: out  = v * attn          (M=p,   N=d,   K=c)
// ---------------------------------------------------------------------------
__global__ void __launch_bounds__(256) k3_window_attn(const float* __restrict__ qkvw,
                                                      const float* __restrict__ temp,
                                                      float* __restrict__ attnO,
                                                      float* __restrict__ outbuf) {
  __shared__ __align__(16) float qk[16384];      // q [0,8192), k [8192,16384)
  __shared__ float scale[128];

  const int tid = threadIdx.x;
  const int bw  = blockIdx.x;
  const int b   = bw >> 10;
  const int win = bw & (NWIN_ - 1);
  const float* base = qkvw + (size_t)((b * NWIN_ + win) * 3) * (P_ * C_);
  const float* vb   = base + 2 * (P_ * C_);      // which=2 -> v[p][c]

  // async q+k fill: 4096 x b128 chunks, 16 per thread
#pragma unroll 4
  for (int i = 0; i < 16; ++i) {
    const int chunk = tid + i * 256;
    unsigned ldsAddr = (unsigned)(size_t)(&qk[chunk * 4]);
    const float* gp  = base + chunk * 4;
    asm volatile("global_load_async_to_lds_b128 %0, %1, off"
                 :: "v"(ldsAddr), "v"(gp) : "memory");
  }
  // prefetch v (32KB = 512 lines) while the async engine fills LDS
  __builtin_prefetch(vb + tid * 16, 0, 3);
  __builtin_prefetch(vb + (tid + 256) * 16, 0, 3);
  asm volatile("s_wait_asynccnt 0" ::: "memory");
  __syncthreads();

  // per-pixel L2 norms (tid<64: q pixel, tid in [64,128): k pixel)
  if (tid < 128) {
    const int p   = tid & 63;
    const int off = (tid < 64 ? 0 : 8192) + p * C_;
    float s = 0.f;
    for (int c = 0; c < C_; ++c) { const float v = qk[off + c]; s += v * v; }
    scale[tid] = 1.0f / fmaxf(sqrtf(s), 1e-12f);
  }
  __syncthreads();
  for (int e = tid; e < 16384; e += 256) {
    const int p = (e >> 7) & 63;
    qk[e] *= scale[(e < 8192 ? 0 : 64) + p];
  }
  __syncthreads();

  const float tval = temp[0];
  const int lane = tid & 31;
  const int wv   = tid >> 5;
  const int mrel = lane & 15;
  const int kb   = (lane >> 4) << 1;
  const int rowHi = (lane >> 4) * 8;

  // ---- GEMM1: attn[c][d] = relu(T * sum_p q[p][c] k[p][d]) ----
  const int m0 = wv * 16;                        // 8 waves x 16 = 128 rows
  const size_t abase = ((size_t)(b * NWIN_ + win)) * C_ * C_;
  float stash[64];
#pragma unroll
  for (int nt = 0; nt < 8; ++nt) {
    const int n0 = nt * 16;
    v8f acc = {};
    for (int kc = 0; kc < P_; kc += 4) {
      v2f a, bf;
      a.x  = qk[(kc + kb) * C_ + m0 + mrel];     // A = q^T : A[m][k]=q[k][m]
      a.y  = qk[(kc + kb + 1) * C_ + m0 + mrel];
      bf.x = qk[8192 + (kc + kb) * C_ + n0 + mrel];
      bf.y = qk[8192 + (kc + kb + 1) * C_ + n0 + mrel];
      acc = wmma4(a, bf, acc);
    }
#pragma unroll
    for (int r = 0; r < 8; ++r) {
      const float v = fmaxf(acc[r] * tval, 0.0f);
      attnO[abase + (size_t)(m0 + rowHi + r) * C_ + n0 + mrel] = v;
      stash[nt * 8 + r] = v;
    }
  }
  __syncthreads();                               // all waves done reading q,k
#pragma unroll
  for (int nt = 0; nt < 8; ++nt)
#pragma unroll
    for (int r = 0; r < 8; ++r)                  // overlay attn[c][d] in LDS
      qk[(m0 + rowHi + r) * C_ + nt * 16 + mrel] = stash[nt * 8 + r];
  __syncthreads();

  // ---- GEMM2: out[p][d] = sum_c v[p][c] attn[c][d] ----
  const int m02    = (wv & 3) * 16;              // 4 pixel tiles
  const int ntBase = (wv >> 2) * 4;              // 2 halves of 8 d-tiles
  float* ob = outbuf + (size_t)((b * NWIN_ + win) * P_) * C_;
  for (int nt = ntBase; nt < ntBase + 4; ++nt) {
    const int n0 = nt * 16;
    v8f acc = {};
    for (int kc = 0; kc < C_; kc += 4) {
      v2f a, bf;
      a.x  = vb[(m02 + mrel) * C_ + kc + kb];    // A[m][k]=v[p=m][c=k]
      a.y  = vb[(m02 + mrel) * C_ + kc + kb + 1];
      bf.x = qk[(kc + kb) * C_ + n0 + mrel];     // B[k][n]=attn[c=k][d=n]
      bf.y = qk[(kc + kb + 1) * C_ + n0 + mrel];
      acc = wmma4(a, bf, acc);
    }
#pragma unroll
    for (int r = 0; r < 8; ++r)
      ob[(size_t)(m02 + rowHi + r) * C_ + n0 + mrel] = acc[r];
  }
}

// ---------------------------------------------------------------------------
// K4: y = W_proj[128x128] @ out(un-windowed, rolled by +4,+4).
// One wave per 16-pixel tile: gather its 64 B-floats once, sweep 8 M-tiles.
// ---------------------------------------------------------------------------
__global__ void __launch_bounds__(256) k4_proj_gemm(const float* __restrict__ outbuf,
                                                    const float* __restrict__ wp,
                                                    float* __restrict__ y) {
  const int tid  = threadIdx.x;
  const int lane = tid & 31;
  const int wv   = tid >> 5;
  const int pt   = blockIdx.x * 8 + wv;          // 16384 pixel tiles
  const int pixBase = pt * 16;
  const int b   = pixBase >> 16;
  const int rem = pixBase & (HW_ - 1);
  const int h   = rem >> 8;
  const int wb  = rem & (W_ - 1);

  const int mrel = lane & 15;
  const int kb   = (lane >> 4) << 1;
  // roll(+4): y-source spatial = (h-4, w-4) mod 256, window-decomposed
  const int hs  = (h - 4) & (H_ - 1);
  const int ws2 = (wb + mrel - 4) & (W_ - 1);
  const int win = (hs >> 3) * 32 + (ws2 >> 3);
  const int p   = (hs & 7) * 8 + (ws2 & 7);
  const float* ob = outbuf + (size_t)((b * NWIN_ + win) * P_ + p) * C_;

  v2f barr[32];
#pragma unroll
  for (int i = 0; i < 32; ++i) {
    barr[i].x = ob[4 * i + kb];
    barr[i].y = ob[4 * i + kb + 1];
  }

  const int rowHi = (lane >> 4) * 8;
  for (int mt = 0; mt < 8; ++mt) {
    const int m0 = mt * 16;
    v8f acc = {};
#pragma unroll
    for (int i = 0; i < 32; ++i) {
      v2f a;
      a.x = wp[(m0 + mrel) * C_ + 4 * i + kb];
      a.y = wp[(m0 + mrel) * C_ + 4 * i + kb + 1];
      acc = wmma4(a, barr[i], acc);
    }
#pragma unroll
    for (int r = 0; r < 8; ++r)
      y[((size_t)(b * C_ + m0 + rowHi + r)) * HW_ + h * W_ + wb + mrel] = acc[r];
  }
}

// ---------------------------------------------------------------------------
extern "C" void kernel_launch(void* const* d_in, const int* in_sizes, int n_in,
                              void* d_out, int out_size, void* d_ws, size_t ws_size,
                              hipStream_t stream) {
  const float* x      = (const float*)d_in[0];   // [4,128,256,256]
  const float* w_qkv  = (const float*)d_in[1];   // [384,128]
  const float* w_dw   = (const float*)d_in[2];   // [384,1,3,3]
  const float* w_proj = (const float*)d_in[3];   // [128,128]
  const float* temp   = (const float*)d_in[4];   // [1]

  const size_t nY = (size_t)B_ * C_ * HW_;             // 33,554,432
  float* y     = (float*)d_out;
  float* attnO = (float*)d_out + nY;                   // [4,1024,128,128]

  const size_t nQKV = (size_t)B_ * C3_ * HW_;          // 100,663,296 floats
  if (ws_size < 2 * nQKV * sizeof(float)) return;      // need ~806 MB scratch
  float* qkv_raw = (float*)d_ws;                       // buffer A
  float* qkvw    = qkv_raw + nQKV;                     // buffer B (windowed)
  float* outbuf  = qkv_raw;                            // reuse A after K2

  // K1: 16384 pixel-tile waves -> 2048 blocks (each wave sweeps 24 M-tiles)
  k1_qkv_gemm<<<2048, 256, 0, stream>>>(x, w_qkv, qkv_raw);
  // K2: one block per (b, q/k/v, window) = 12288
  k2_dwconv<<<12288, 256, 0, stream>>>(qkv_raw, w_dw, qkvw);
  // K3: one block per window: 4 * 1024
  k3_window_attn<<<4096, 256, 0, stream>>>(qkvw, temp, attnO, outbuf);
  // K4: 16384 pixel-tile waves -> 2048 blocks (each wave sweeps 8 M-tiles)
  k4_proj_gemm<<<2048, 256, 0, stream>>>(outbuf, w_proj, y);
}